// MYIN_57543971832335
// MI455X (gfx1250) — compile-verified
//
#include <hip/hip_runtime.h>
#include <math.h>

// ---------------- problem constants ----------------
#define HW_H   256
#define HW_W   256
#define HWPIX  (HW_H * HW_W)          // 65536
#define PSIZE  16
#define PSTRIDE 8
#define NUMS   31                     // (256-16)/8+1
#define LREAL  961                    // 31*31
#define LPAD   976                    // 61*16
#define LTILES 61
#define BB     2
#define CC     64
#define EPSV   1e-5f
#define NBIAS  3721                   // (2*31-1)^2

// ---------------- workspace layout (floats) ----------------
// q, k stored TRANSPOSED: (B, LPAD, C)  -> WMMA operand pairs contiguous
// mu1, var1, patch_back, patch_std: (B, C, LPAD)
#define OF_MEANF 0
#define OF_STDF  (OF_MEANF + BB*CC)
#define OF_MEANB (OF_STDF  + BB*CC)
#define OF_STDB  (OF_MEANB + BB*CC)
#define OF_Q     (OF_STDB  + BB*CC)
#define OF_K     (OF_Q    + BB*CC*LPAD)
#define OF_MU1   (OF_K    + BB*CC*LPAD)
#define OF_VAR1  (OF_MU1  + BB*CC*LPAD)
#define OF_PB    (OF_VAR1 + BB*CC*LPAD)
#define OF_PS    (OF_PB   + BB*CC*LPAD)
#define OF_S     (OF_PS   + BB*CC*LPAD)          // B*LPAD*LPAD

typedef __attribute__((ext_vector_type(2))) float v2f;
typedef __attribute__((ext_vector_type(8))) float v8f;

__device__ __forceinline__ float waveSum(float v) {
    #pragma unroll
    for (int o = 16; o > 0; o >>= 1) v += __shfl_down(v, o, 32);
    return v;
}
__device__ __forceinline__ float waveMax(float v) {
    #pragma unroll
    for (int o = 16; o > 0; o >>= 1) v = fmaxf(v, __shfl_down(v, o, 32));
    return v;
}

// ============================================================
// Kernel 1: global masked stats per (b,c): mean/std for fg and bg
// ============================================================
__global__ void k_global_stats(const float* __restrict__ x,
                               const float* __restrict__ mask,
                               float* __restrict__ ws) {
    const int bc = blockIdx.x;          // b*C + c
    const int b  = bc >> 6;
    const float* xp = x + (size_t)bc * HWPIX;
    const float* mp = mask + (size_t)b * HWPIX;

    float s1f = 0.f, s2f = 0.f, nf = 0.f, s1b = 0.f, s2b = 0.f, nb = 0.f;
    for (int i = threadIdx.x; i < HWPIX; i += 256) {
        float xv = xp[i];
        float m  = mp[i];
        float om = 1.0f - m;
        s1f += m * xv;  s2f += m * xv * xv;  nf += m;
        s1b += om * xv; s2b += om * xv * xv; nb += om;
    }
    __shared__ float sm[8][6];
    const int wid = threadIdx.x >> 5, lane = threadIdx.x & 31;
    s1f = waveSum(s1f); s2f = waveSum(s2f); nf = waveSum(nf);
    s1b = waveSum(s1b); s2b = waveSum(s2b); nb = waveSum(nb);
    if (lane == 0) {
        sm[wid][0] = s1f; sm[wid][1] = s2f; sm[wid][2] = nf;
        sm[wid][3] = s1b; sm[wid][4] = s2b; sm[wid][5] = nb;
    }
    __syncthreads();
    if (threadIdx.x == 0) {
        float t[6] = {0.f, 0.f, 0.f, 0.f, 0.f, 0.f};
        for (int w = 0; w < 8; ++w)
            for (int j = 0; j < 6; ++j) t[j] += sm[w][j];
        float muf = t[0] / (t[2] + EPSV);
        float vrf = (t[1] - 2.f * muf * t[0] + muf * muf * t[2]) / (t[2] + EPSV);
        float mub = t[3] / (t[5] + EPSV);
        float vrb = (t[4] - 2.f * mub * t[3] + mub * mub * t[5]) / (t[5] + EPSV);
        ws[OF_MEANF + bc] = muf;
        ws[OF_STDF  + bc] = sqrtf(vrf + EPSV);
        ws[OF_MEANB + bc] = mub;
        ws[OF_STDB  + bc] = sqrtf(vrb + EPSV);
    }
}

// ============================================================
// Kernel 2: per-patch stats. q,k -> (B,LPAD,C); mu1,var1 -> (B,C,LPAD)
// one wave per (b,c,lp)
// ============================================================
__global__ void k_patch_stats(const float* __restrict__ x,
                              const float* __restrict__ mask,
                              const float* __restrict__ qg, const float* __restrict__ qb,
                              const float* __restrict__ kg, const float* __restrict__ kb_,
                              float* __restrict__ ws) {
    const int wid  = threadIdx.x >> 5;
    const int lane = threadIdx.x & 31;
    const int job  = blockIdx.x * 8 + wid;          // B*C*LPAD jobs
    const int b    = job / (CC * LPAD);
    const int rem  = job % (CC * LPAD);
    const int c    = rem / LPAD;
    const int lp   = rem % LPAD;
    const int qkidx = (b * LPAD + lp) * CC + c;     // transposed layout
    const int mvidx = (b * CC + c) * LPAD + lp;

    if (lp >= LREAL) {
        if (lane == 0) {
            ws[OF_Q + qkidx] = 0.f; ws[OF_K + qkidx] = 0.f;
            ws[OF_MU1 + mvidx] = 0.f; ws[OF_VAR1 + mvidx] = 0.f;
        }
        return;
    }
    const int pi = lp / NUMS, pj = lp % NUMS;
    const int r0 = pi * PSTRIDE, c0 = pj * PSTRIDE;
    const float* xp = x + (size_t)(b * CC + c) * HWPIX;
    const float* mp = mask + (size_t)b * HWPIX;

    float s1b = 0.f, s2b = 0.f, s1f = 0.f, nb = 0.f, nfc = 0.f;
    #pragma unroll
    for (int t = 0; t < 8; ++t) {
        int p  = lane + t * 32;                 // 0..255
        int pr = p >> 4, pc = p & 15;
        int gidx = (r0 + pr) * HW_W + (c0 + pc);
        float xv = xp[gidx];
        float m  = mp[gidx];
        float om = 1.0f - m;
        s1b += om * xv; s2b += om * xv * xv; nb += om;
        s1f += m * xv;  nfc += m;
    }
    s1b = waveSum(s1b); s2b = waveSum(s2b); s1f = waveSum(s1f);
    nb  = waveSum(nb);  nfc = waveSum(nfc);

    if (lane == 0) {
        const int bc = b * CC + c;
        float mean_b = ws[OF_MEANB + bc], std_b = ws[OF_STDB + bc];
        float mean_f = ws[OF_MEANF + bc], std_f = ws[OF_STDF + bc];
        float mu1v = s1b / (nb + EPSV);
        float var1 = (s2b - 2.f * mu1v * s1b + mu1v * mu1v * nb) / (nb + EPSV);
        float mu4  = ((s1b - mean_b * nb) / std_b) / (nb + EPSV);
        float mu3  = ((s1f - mean_f * nfc) / std_f) / (nfc + EPSV);
        ws[OF_Q   + qkidx] = mu3 * qg[c] + qb[c];
        ws[OF_K   + qkidx] = mu4 * kg[c] + kb_[c];
        ws[OF_MU1 + mvidx] = mu1v;
        ws[OF_VAR1+ mvidx] = var1;
    }
}

// ============================================================
// Kernel 3a: logits = q^T k + bias(LDS), row softmax -> S (B,LPAD,LPAD)
// grid = (61 l-tiles, B), block 256 (8 waves). WMMA f32 16x16x4.
// A operands hoisted to registers; all operand loads are b64.
// ============================================================
__global__ void k_attn_softmax(const float* __restrict__ ws_ro,
                               const float* __restrict__ bias_table,
                               float* __restrict__ ws) {
    const float* q = ws_ro + OF_Q;      // (B, LPAD, C)
    const float* k = ws_ro + OF_K;
    float* S = ws + OF_S;

    const int lt = blockIdx.x, b = blockIdx.y;
    const int l0 = lt * 16;
    const int wid  = threadIdx.x >> 5;
    const int lane = threadIdx.x & 31;
    const int lm   = lane & 15;
    const int half = lane >> 4;
    const int kb   = half << 1;                 // K base: 0 or 2

    __shared__ float strip[16][LPAD];           // 62464 B
    __shared__ float biasLds[NBIAS];            // 14884 B

    for (int i = threadIdx.x; i < NBIAS; i += 256) biasLds[i] = bias_table[i];

    // hoist A tile (invariant across m-tiles): 16 x b64 loads
    v2f areg[16];
    const float* qbase = q + ((size_t)b * LPAD + l0 + lm) * CC + kb;
    #pragma unroll
    for (int i = 0; i < 16; ++i)
        areg[i] = *reinterpret_cast<const v2f*>(qbase + i * 4);

    __syncthreads();

    for (int mt = wid; mt < LTILES; mt += 8) {
        const int m0 = mt * 16;
        const float* kbase = k + ((size_t)b * LPAD + m0 + lm) * CC + kb;
        v8f acc = {};
        #pragma unroll
        for (int i = 0; i < 16; ++i) {
            v2f bv = *reinterpret_cast<const v2f*>(kbase + i * 4);
            acc = __builtin_amdgcn_wmma_f32_16x16x4_f32(
                false, areg[i], false, bv, (short)0, acc, false, false);
        }
        const int mcol = m0 + lm;
        const int im = mcol / NUMS, jm = mcol - im * NUMS;
        #pragma unroll
        for (int r = 0; r < 8; ++r) {
            const int M = r + 8 * half;
            const int l = l0 + M;
            int il = l / NUMS, jl = l - il * NUMS;
            int bidx = (il - im + NUMS - 1) * (2 * NUMS - 1) + (jl - jm + NUMS - 1);
            bidx = min(max(bidx, 0), NBIAS - 1);            // branchless clamp
            float v = acc[r] + biasLds[bidx];
            if (l >= LREAL || mcol >= LREAL) v = -3.0e38f;  // pad -> -inf
            strip[M][mcol] = v;
        }
    }
    __syncthreads();

    // softmax: each wave owns 2 rows
    for (int rr = 2 * wid; rr < 2 * wid + 2; ++rr) {
        const int l = l0 + rr;
        float* Srow = S + ((size_t)b * LPAD + l) * LPAD;
        if (l >= LREAL) {
            for (int m = lane; m < LPAD; m += 32) Srow[m] = 0.f;
            continue;
        }
        float mx = -3.0e38f;
        for (int m = lane; m < LREAL; m += 32) mx = fmaxf(mx, strip[rr][m]);
        mx = waveMax(mx);
        mx = __shfl(mx, 0, 32);
        float sum = 0.f;
        for (int m = lane; m < LREAL; m += 32) {
            float e = expf(strip[rr][m] - mx);
            strip[rr][m] = e;
            sum += e;
        }
        sum = waveSum(sum);
        sum = __shfl(sum, 0, 32);
        const float inv = 1.0f / sum;
        for (int m = lane; m < LREAL; m += 32) Srow[m] = strip[rr][m] * inv;
        for (int m = LREAL + lane; m < LPAD; m += 32) Srow[m] = 0.f;
    }
}

// ============================================================
// Kernel 3b: patch_back = S*mu1^T, patch_std = S*var1^T (per b)
// one wave per (b, l-tile, c-tile); 488 jobs = 61 blocks * 8 waves
// All operand loads are contiguous b64.
// ============================================================
__global__ void k_attn_apply(const float* __restrict__ ws_ro,
                             float* __restrict__ ws) {
    const float* S    = ws_ro + OF_S;
    const float* mu1  = ws_ro + OF_MU1;
    const float* var1 = ws_ro + OF_VAR1;

    const int wid  = threadIdx.x >> 5;
    const int lane = threadIdx.x & 31;
    const int job  = blockIdx.x * 8 + wid;      // 0..487
    const int b    = job / (LTILES * 4);
    const int rem  = job % (LTILES * 4);
    const int lt   = rem >> 2;
    const int ct   = rem & 3;
    const int l0   = lt * 16;
    const int c0   = ct * 16;

    const int lm   = lane & 15;
    const int half = lane >> 4;
    const int kb   = half << 1;

    v8f accB = {};
    v8f accS = {};
    const float* Sbase  = S    + ((size_t)b * LPAD + l0 + lm) * LPAD + kb;
    const float* m1base = mu1  + (size_t)(b * CC + c0 + lm) * LPAD + kb;
    const float* v1base = var1 + (size_t)(b * CC + c0 + lm) * LPAD + kb;

    #pragma unroll 4
    for (int mm = 0; mm < LPAD; mm += 4) {
        v2f a  = *reinterpret_cast<const v2f*>(Sbase  + mm);
        v2f b1 = *reinterpret_cast<const v2f*>(m1base + mm);
        v2f b2 = *reinterpret_cast<const v2f*>(v1base + mm);
        accB = __builtin_amdgcn_wmma_f32_16x16x4_f32(
            false, a, false, b1, (short)0, accB, false, false);
        accS = __builtin_amdgcn_wmma_f32_16x16x4_f32(
            false, a, false, b2, (short)0, accS, false, false);
    }
    #pragma unroll
    for (int r = 0; r < 8; ++r) {
        const int M = r + 8 * half;
        const int l = l0 + M;
        if (l < LREAL) {
            const int oidx = (b * CC + c0 + lm) * LPAD + l;
            ws[OF_PB + oidx] = accB[r];
            ws[OF_PS + oidx] = accS[r];
        }
    }
}

// ============================================================
// Kernel 4: analytic fold + final affine/compose. One thread/pixel.
// ============================================================
__global__ void k_final(const float* __restrict__ x,
                        const float* __restrict__ mask,
                        const float* __restrict__ fg_g, const float* __restrict__ fg_b,
                        const float* __restrict__ bg_g, const float* __restrict__ bg_b,
                        const float* __restrict__ ws,
                        float* __restrict__ out) {
    const int idx = blockIdx.x * 256 + threadIdx.x;   // B*C*H*W
    const int b   = idx / (CC * HWPIX);
    const int c   = (idx / HWPIX) % CC;
    const int pix = idx % HWPIX;
    const int h   = pix / HW_W;
    const int w   = pix % HW_W;
    const int bc  = b * CC + c;

    const float xv = x[idx];
    const float m  = mask[b * HWPIX + pix];
    const float om = 1.0f - m;

    const float norm_bg = (xv * (1.0f + bg_g[c]) + bg_b[c]) * om;
    const float norm_fg = (xv - ws[OF_MEANF + bc]) / ws[OF_STDF + bc] * m;

    int ilo = (h - 8) / 8; if (ilo < 0) ilo = 0;
    int ihi = h / 8;       if (ihi > NUMS - 1) ihi = NUMS - 1;
    int jlo = (w - 8) / 8; if (jlo < 0) jlo = 0;
    int jhi = w / 8;       if (jhi > NUMS - 1) jhi = NUMS - 1;

    float sb = 0.f, ss = 0.f;
    int cnt = 0;
    const float* pbp = ws + OF_PB + (size_t)bc * LPAD;
    const float* psp = ws + OF_PS + (size_t)bc * LPAD;
    for (int i = ilo; i <= ihi; ++i) {
        for (int j = jlo; j <= jhi; ++j) {
            const int l = i * NUMS + j;
            sb += pbp[l];
            ss += psp[l];
            ++cnt;
        }
    }
    const float nfv = (norm_fg * ss + sb) / (float)cnt;
    const float nf  = (nfv * (1.0f + fg_g[c]) + fg_b[c]) * m;
    out[idx] = nf + norm_bg;
}

// ============================================================
extern "C" void kernel_launch(void* const* d_in, const int* in_sizes, int n_in,
                              void* d_out, int out_size, void* d_ws, size_t ws_size,
                              hipStream_t stream) {
    const float* x    = (const float*)d_in[0];
    const float* mask = (const float*)d_in[1];
    const float* fg_g = (const float*)d_in[2];
    const float* fg_b = (const float*)d_in[3];
    const float* bg_g = (const float*)d_in[4];
    const float* bg_b = (const float*)d_in[5];
    const float* q_g  = (const float*)d_in[6];
    const float* q_b  = (const float*)d_in[7];
    const float* k_g  = (const float*)d_in[8];
    const float* k_b  = (const float*)d_in[9];
    const float* bias = (const float*)d_in[10];
    float* out = (float*)d_out;
    float* ws  = (float*)d_ws;

    k_global_stats<<<dim3(BB * CC), dim3(256), 0, stream>>>(x, mask, ws);
    k_patch_stats<<<dim3(BB * CC * LPAD / 8), dim3(256), 0, stream>>>(
        x, mask, q_g, q_b, k_g, k_b, ws);
    k_attn_softmax<<<dim3(LTILES, BB), dim3(256), 0, stream>>>(ws, bias, ws);
    k_attn_apply<<<dim3(LTILES), dim3(256), 0, stream>>>(ws, ws);
    k_final<<<dim3(BB * CC * HWPIX / 256), dim3(256), 0, stream>>>(
        x, mask, fg_g, fg_b, bg_g, bg_b, ws, out);
}